// KAReader_13340168421496
// MI455X (gfx1250) — compile-verified
//
#include <hip/hip_runtime.h>
#include <hip/hip_bf16.h>
#include <math.h>

#define Bq 16
#define Tq 20
#define Cc 512
#define NBn 50
#define Rr 500
#define TRr 10
#define NWw 50000
#define NEe 100000
#define WDd 300
#define EDd 100
#define Hh 64
#define HP 66   // padded LDS row pitch: (2r+k) mod 64 banks -> conflict-free, keeps 8B align
#define HC 32
#define G4 128
#define BC (Bq*Cc)

typedef __attribute__((ext_vector_type(2))) float v2f;
typedef __attribute__((ext_vector_type(8))) float v8f;

__device__ __forceinline__ float sigm(float x){ return 1.0f/(1.0f+__expf(-x)); }
__device__ __forceinline__ float lrelu(float x){ return x > 0.0f ? x : 0.01f*x; }

__device__ __forceinline__ v8f wmma_f32_4(v2f a, v2f b, v8f c){
  return __builtin_amdgcn_wmma_f32_16x16x4_f32(false, a, false, b, (short)0, c, false, false);
}

// ---------------------------------------------------------------------------
// BiLSTM: one block per sequence, 128 threads. Replicates masked_lstm + bilstm.
// out: (N, Tn, 64) = [h_fwd(32) | h_bwd_gathered(32)]
// ---------------------------------------------------------------------------
__global__ void bilstm_kernel(const int* __restrict__ tokens, int N, int Tn,
                              const float* __restrict__ word_emb,
                              const float* __restrict__ Wih_f, const float* __restrict__ Whh_f, const float* __restrict__ b_f,
                              const float* __restrict__ Wih_b, const float* __restrict__ Whh_b, const float* __restrict__ b_b,
                              float* __restrict__ out, int minlen)
{
  __shared__ float s_x[WDd];
  __shared__ float s_h[HC];
  __shared__ float s_c[HC];
  __shared__ float s_g[G4];
  __shared__ float s_y[Tq][HC];
  __shared__ int s_cnt;

  const int n = blockIdx.x;
  const int tid = threadIdx.x;

  if (tid == 0) s_cnt = 0;
  __syncthreads();
  if (tid < Tn) { if (tokens[n*Tn + tid] != 1) atomicAdd(&s_cnt, 1); }
  __syncthreads();
  const int len = max(s_cnt, minlen);

  // ---------------- forward direction ----------------
  if (tid < HC) { s_h[tid] = 0.0f; s_c[tid] = 0.0f; }
  __syncthreads();
  for (int t = 0; t < Tn; ++t) {
    const int tok = tokens[n*Tn + t];
    for (int k = tid; k < WDd; k += 128) s_x[k] = word_emb[(size_t)tok*WDd + k];
    __syncthreads();
    float g = b_f[tid];
    for (int k = 0; k < WDd; ++k) g += s_x[k] * Wih_f[k*G4 + tid];
    for (int k = 0; k < HC;  ++k) g += s_h[k] * Whh_f[k*G4 + tid];
    s_g[tid] = g;
    __syncthreads();
    if (tid < HC) {
      float ig = sigm(s_g[tid]);
      float fg = sigm(s_g[HC + tid]);
      float gg = tanhf(s_g[2*HC + tid]);
      float og = sigm(s_g[3*HC + tid]);
      float c2 = fg * s_c[tid] + ig * gg;
      float h2 = og * tanhf(c2);
      float m = (t < len) ? 1.0f : 0.0f;
      s_h[tid] = m*h2 + (1.0f-m)*s_h[tid];
      s_c[tid] = m*c2 + (1.0f-m)*s_c[tid];
      out[((size_t)n*Tn + t)*(2*HC) + tid] = s_h[tid]*m;
    }
    __syncthreads();
  }

  // ---------------- backward direction (reversed input) ----------------
  if (tid < HC) { s_h[tid] = 0.0f; s_c[tid] = 0.0f; }
  __syncthreads();
  for (int t = 0; t < Tn; ++t) {
    int src = len - 1 - t; src = min(max(src, 0), Tn - 1);
    const int tok = tokens[n*Tn + src];
    const float vmul = (t < len) ? 1.0f : 0.0f;
    for (int k = tid; k < WDd; k += 128) s_x[k] = word_emb[(size_t)tok*WDd + k] * vmul;
    __syncthreads();
    float g = b_b[tid];
    for (int k = 0; k < WDd; ++k) g += s_x[k] * Wih_b[k*G4 + tid];
    for (int k = 0; k < HC;  ++k) g += s_h[k] * Whh_b[k*G4 + tid];
    s_g[tid] = g;
    __syncthreads();
    if (tid < HC) {
      float ig = sigm(s_g[tid]);
      float fg = sigm(s_g[HC + tid]);
      float gg = tanhf(s_g[2*HC + tid]);
      float og = sigm(s_g[3*HC + tid]);
      float c2 = fg * s_c[tid] + ig * gg;
      float h2 = og * tanhf(c2);
      float m = (t < len) ? 1.0f : 0.0f;
      s_h[tid] = m*h2 + (1.0f-m)*s_h[tid];
      s_c[tid] = m*c2 + (1.0f-m)*s_c[tid];
      s_y[t][tid] = s_h[tid]*m;
    }
    __syncthreads();
  }
  // gather back: hb[t] = hr[clip(len-1-t)] * valid(t)
  for (int e = tid; e < Tn*HC; e += 128) {
    int t = e / HC, j = e % HC;
    int src = len - 1 - t; src = min(max(src, 0), Tn - 1);
    float v = (t < len) ? s_y[src][j] : 0.0f;
    out[((size_t)n*Tn + t)*(2*HC) + HC + j] = v;
  }
}

// ---------------------------------------------------------------------------
// Attention pooling over rel encodings: block per r, 64 threads
// ---------------------------------------------------------------------------
__global__ void attnpool_kernel(const float* __restrict__ seq,   // (R,TR,64)
                                const int* __restrict__ rel_word_ids,
                                const float* __restrict__ att_w,
                                const float* __restrict__ att_b,
                                float* __restrict__ outp)        // (R,64)
{
  __shared__ float s_p[TRr];
  const int r = blockIdx.x;
  const int tid = threadIdx.x;
  if (tid < TRr) {
    float s = att_b[0];
    const float* xr = seq + ((size_t)r*TRr + tid)*Hh;
    for (int h = 0; h < Hh; ++h) s += xr[h]*att_w[h];
    float m = (rel_word_ids[r*TRr + tid] != 1) ? 1.0f : 0.0f;
    s_p[tid] = s - (1.0f - m)*1e20f;
  }
  __syncthreads();
  if (tid == 0) {
    float mx = -3e38f;
    for (int t = 0; t < TRr; ++t) mx = fmaxf(mx, s_p[t]);
    float sum = 0.0f;
    for (int t = 0; t < TRr; ++t) { float e = __expf(s_p[t]-mx); s_p[t] = e; sum += e; }
    float inv = 1.0f/sum;
    for (int t = 0; t < TRr; ++t) s_p[t] *= inv;
  }
  __syncthreads();
  float acc = 0.0f;
  for (int t = 0; t < TRr; ++t) acc += s_p[t]*seq[((size_t)r*TRr + t)*Hh + tid];
  outp[(size_t)r*Hh + tid] = acc;
}

// ---------------------------------------------------------------------------
// ent0 = lrelu(entity_table[cand] @ W_ent + b_ent) via WMMA f32 16x16x4
// grid: 512 blocks * 128 threads (4 waves, one 16x16 N-tile each)
// ---------------------------------------------------------------------------
__global__ void ent0_kernel(const int* __restrict__ cand,
                            const float* __restrict__ entity_table,
                            const float* __restrict__ W_ent,
                            const float* __restrict__ b_ent,
                            float* __restrict__ ent0)
{
  __shared__ float s_a[16][EDd];   // pitch 100: 36r mod 64 -> 16 distinct banks
  __shared__ float s_b[EDd][Hh];
  const int tid = threadIdx.x;
  const int m0 = blockIdx.x * 16;
  for (int e = tid; e < 16*EDd; e += 128) {
    int i = e / EDd, k = e % EDd;
    int ent = cand[m0 + i];
    s_a[i][k] = entity_table[(size_t)ent*EDd + k];
  }
  for (int e = tid; e < EDd*Hh; e += 128) {
    int k = e / Hh, h = e % Hh;
    s_b[k][h] = W_ent[k*Hh + h];
  }
  __syncthreads();
  const int wave = tid >> 5, lane = tid & 31, half = lane >> 4, r = lane & 15;
  const int n0 = wave * 16;
  v8f acc = {};
  for (int k = 0; k < EDd; k += 4) {
    int kk = k + 2*half;
    v2f a; a.x = s_a[r][kk]; a.y = s_a[r][kk+1];
    v2f b; b.x = s_b[kk][n0+r]; b.y = s_b[kk+1][n0+r];
    acc = wmma_f32_4(a, b, acc);
  }
  for (int j = 0; j < 8; ++j) {
    int row = j + 8*half, col = n0 + r;
    ent0[(size_t)(m0 + row)*Hh + col] = lrelu(acc[j] + b_ent[col]);
  }
}

// ---------------------------------------------------------------------------
// Main per-candidate kernel: block per bc (8192 blocks), 256 threads (8 waves)
// ---------------------------------------------------------------------------
__global__ void __launch_bounds__(256)
main_kernel(const int* __restrict__ questions,
            const int* __restrict__ link_ents,
            const int* __restrict__ link_rels,
            const float* __restrict__ q_emb,     // (B,T,64)
            const float* __restrict__ rel_pool,  // (R,64)
            const float* __restrict__ ent0,      // (BC,64)
            const float* __restrict__ W_qrel, const float* __restrict__ b_qrel,
            const float* __restrict__ W_cq,   const float* __restrict__ b_cq,
            float* __restrict__ ent1,            // (BC,64)
            float* __restrict__ qsimi)           // (BC,50)
{
  __shared__ float s_qe[32][HP];
  __shared__ float s_nbrel[64][HP];
  __shared__ float s_aff[32][HP];   // reused as rel_aware
  __shared__ float s_pq[32][HP];
  __shared__ float s_pn[32][HP];
  __shared__ float s_rq[64][HP];
  __shared__ float s_rr[32][HP];
  __shared__ float s_qm[32];
  __shared__ float s_nm[64];
  __shared__ float s_qnode[Hh];

  const int bc = blockIdx.x;
  const int b = bc / Cc;
  const int tid = threadIdx.x;
  const int wave = tid >> 5, lane = tid & 31, half = lane >> 4, r = lane & 15;

  // loads with zero padding
  for (int e = tid; e < 32*Hh; e += 256) {
    int t = e / Hh, h = e % Hh;
    s_qe[t][h] = (t < Tq) ? q_emb[((size_t)b*Tq + t)*Hh + h] : 0.0f;
  }
  if (tid < 32) s_qm[tid] = (tid < Tq && questions[b*Tq + tid] != 1) ? 1.0f : 0.0f;
  for (int e = tid; e < 64*Hh; e += 256) {
    int n = e / Hh, h = e % Hh;
    if (n < NBn) {
      int rel = link_rels[(size_t)bc*NBn + n];
      s_nbrel[n][h] = rel_pool[(size_t)rel*Hh + h];
    } else s_nbrel[n][h] = 0.0f;
  }
  if (tid < 64) s_nm[tid] = (tid < NBn && link_ents[(size_t)bc*NBn + tid] != NEe) ? 1.0f : 0.0f;
  __syncthreads();

  // GEMM1: aff[t][n] = sum_h qe[t][h]*nbrel[n][h]   (M=32,N=64,K=64) 8 tiles/8 waves
  {
    int m0 = (wave >> 2)*16, n0 = (wave & 3)*16;
    v8f acc = {};
    for (int k = 0; k < Hh; k += 4) {
      int kk = k + 2*half;
      v2f a; a.x = s_qe[m0+r][kk];    a.y = s_qe[m0+r][kk+1];
      v2f bb; bb.x = s_nbrel[n0+r][kk]; bb.y = s_nbrel[n0+r][kk+1];
      acc = wmma_f32_4(a, bb, acc);
    }
    for (int j = 0; j < 8; ++j) s_aff[m0 + j + 8*half][n0 + r] = acc[j];
  }
  __syncthreads();

  // pq: softmax over t (masked with q_mask*1e20); pn: softmax over n<50 (minus (1-nm))
  if (tid < 64) {
    int n = tid;
    float mx = -3e38f;
    for (int t = 0; t < 32; ++t) mx = fmaxf(mx, s_aff[t][n] - (1.0f - s_qm[t])*1e20f);
    float sum = 0.0f;
    for (int t = 0; t < 32; ++t) sum += __expf(s_aff[t][n] - (1.0f - s_qm[t])*1e20f - mx);
    float inv = 1.0f/sum;
    for (int t = 0; t < 32; ++t) s_pq[t][n] = __expf(s_aff[t][n] - (1.0f - s_qm[t])*1e20f - mx)*inv;
  }
  if (tid < 32) {
    int t = tid;
    float mx = -3e38f;
    for (int n = 0; n < NBn; ++n) mx = fmaxf(mx, s_aff[t][n] - (1.0f - s_nm[n]));
    float sum = 0.0f;
    for (int n = 0; n < NBn; ++n) sum += __expf(s_aff[t][n] - (1.0f - s_nm[n]) - mx);
    float inv = 1.0f/sum;
    for (int n = 0; n < 64; ++n)
      s_pn[t][n] = (n < NBn) ? __expf(s_aff[t][n] - (1.0f - s_nm[n]) - mx)*inv : 0.0f;
  }
  __syncthreads();

  // GEMM2: rq[n][h] = sum_t pq[t][n]*qe[t][h]  (M=64,N=64,K=32) 16 tiles, 2/wave
  for (int tile = wave; tile < 16; tile += 8) {
    int m0 = (tile >> 2)*16, n0 = (tile & 3)*16;
    v8f acc = {};
    for (int k = 0; k < 32; k += 4) {
      int kk = k + 2*half;
      v2f a; a.x = s_pq[kk][m0+r];  a.y = s_pq[kk+1][m0+r];
      v2f bb; bb.x = s_qe[kk][n0+r]; bb.y = s_qe[kk+1][n0+r];
      acc = wmma_f32_4(a, bb, acc);
    }
    for (int j = 0; j < 8; ++j) s_rq[m0 + j + 8*half][n0 + r] = acc[j];
  }
  __syncthreads();

  // q_rel_simi
  if (tid < NBn) {
    float s = 0.0f;
    for (int h = 0; h < Hh; ++h) s += s_nbrel[tid][h]*s_rq[tid][h];
    qsimi[(size_t)bc*NBn + tid] = s;
  }

  // GEMM3: rr[t][h] = sum_n pn[t][n]*nbrel[n][h]  (M=32,N=64,K=64)
  {
    int m0 = (wave >> 2)*16, n0 = (wave & 3)*16;
    v8f acc = {};
    for (int k = 0; k < 64; k += 4) {
      int kk = k + 2*half;
      v2f a; a.x = s_pn[m0+r][kk];   a.y = s_pn[m0+r][kk+1];
      v2f bb; bb.x = s_nbrel[kk][n0+r]; bb.y = s_nbrel[kk+1][n0+r];
      acc = wmma_f32_4(a, bb, acc);
    }
    for (int j = 0; j < 8; ++j) s_rr[m0 + j + 8*half][n0 + r] = acc[j];
  }
  __syncthreads();

  // GEMM4: rel_aware[t][h] = tanh([qe|rr](32x128) @ W_qrel(128x64) + b_qrel)
  {
    int m0 = (wave >> 2)*16, n0 = (wave & 3)*16;
    v8f acc = {};
    for (int k = 0; k < 128; k += 4) {
      int kk = k + 2*half;
      v2f a;
      if (k < 64) { a.x = s_qe[m0+r][kk];    a.y = s_qe[m0+r][kk+1]; }
      else        { a.x = s_rr[m0+r][kk-64]; a.y = s_rr[m0+r][kk-63]; }
      v2f bb; bb.x = W_qrel[kk*Hh + n0 + r]; bb.y = W_qrel[(kk+1)*Hh + n0 + r];
      acc = wmma_f32_4(a, bb, acc);
    }
    for (int j = 0; j < 8; ++j)
      s_aff[m0 + j + 8*half][n0 + r] = tanhf(acc[j] + b_qrel[n0 + r]);
  }
  __syncthreads();

  // q_node = max over t<20
  if (tid < Hh) {
    float mx = -3e38f;
    for (int t = 0; t < Tq; ++t) mx = fmaxf(mx, s_aff[t][tid]);
    s_qnode[tid] = mx;
  }
  __syncthreads();

  // ent1 = lrelu([ent0 | q_node] @ W_cq + b_cq)
  if (tid < Hh) {
    float acc = b_cq[tid];
    const float* e0 = ent0 + (size_t)bc*Hh;
    for (int k = 0; k < Hh; ++k) acc += e0[k]*W_cq[k*Hh + tid];
    for (int k = 0; k < Hh; ++k) acc += s_qnode[k]*W_cq[(Hh + k)*Hh + tid];
    ent1[(size_t)bc*Hh + tid] = lrelu(acc);
  }
}

// ---------------------------------------------------------------------------
// Pass 2: neighbor aggregation + gate. Block per bc, 128 threads.
// ---------------------------------------------------------------------------
__global__ void pass2_kernel(const int* __restrict__ link_ents,
                             const int* __restrict__ link_rels,
                             const int* __restrict__ query_entities,
                             const float* __restrict__ rel_pool,
                             const float* __restrict__ ent1,
                             const float* __restrict__ qsimi,
                             const float* __restrict__ W_kgp, const float* __restrict__ b_kgp,
                             const float* __restrict__ W_kgg, const float* __restrict__ b_kgg,
                             const float* __restrict__ W_sp,  const float* __restrict__ b_sp,
                             float* __restrict__ outp)
{
  __shared__ float s_attn[NBn];
  __shared__ int   s_idx[NBn];
  __shared__ int   s_rel[NBn];
  __shared__ float s_w[128];
  __shared__ float s_agg[Hh];
  __shared__ float s_e1[Hh];

  const int bc = blockIdx.x;
  const int b = bc / Cc;
  const int tid = threadIdx.x;

  if (tid < NBn) {
    int ent = link_ents[(size_t)bc*NBn + tid];
    int ml = (ent != -1) ? 1 : 0;
    int idx = (ent + b*Cc + 1)*ml;
    s_idx[tid] = idx;
    s_rel[tid] = link_rels[(size_t)bc*NBn + tid];
    float nm = (ent != NEe) ? 1.0f : 0.0f;
    float seed = (idx == 0) ? 0.0f : (float)query_entities[idx - 1];
    s_attn[tid] = qsimi[(size_t)bc*NBn + tid]*seed - (1.0f - nm)*1e8f;
  }
  if (tid < Hh) s_e1[tid] = ent1[(size_t)bc*Hh + tid];
  __syncthreads();
  if (tid == 0) {
    float mx = -3e38f;
    for (int n = 0; n < NBn; ++n) mx = fmaxf(mx, s_attn[n]);
    float sum = 0.0f;
    for (int n = 0; n < NBn; ++n) { float e = __expf(s_attn[n]-mx); s_attn[n] = e; sum += e; }
    float inv = 1.0f/sum;
    for (int n = 0; n < NBn; ++n) s_attn[n] *= inv;
  }
  __syncthreads();

  // weighted neighbor vector (128): [rel | ent]
  {
    float acc = 0.0f;
    for (int n = 0; n < NBn; ++n) {
      float v;
      if (tid < Hh) v = rel_pool[(size_t)s_rel[n]*Hh + tid];
      else {
        int idx = s_idx[n];
        v = (idx == 0) ? 0.0f : ent1[(size_t)(idx-1)*Hh + (tid - Hh)];
      }
      acc += s_attn[n]*v;
    }
    s_w[tid] = acc;
  }
  __syncthreads();

  if (tid < Hh) {
    float a = b_kgp[tid];
    for (int k = 0; k < 128; ++k) a += s_w[k]*W_kgp[k*Hh + tid];
    s_agg[tid] = lrelu(a);
  }
  __syncthreads();

  if (tid < Hh) {
    float sp = b_sp[tid];
    for (int k = 0; k < Hh; ++k) sp += s_e1[k]*W_sp[k*Hh + tid];
    sp = lrelu(sp);
    float g = b_kgg[tid];
    for (int k = 0; k < Hh; ++k) g += s_agg[k]*W_kgg[k*Hh + tid];
    for (int k = 0; k < Hh; ++k) g += s_e1[k]*W_kgg[(Hh + k)*Hh + tid];
    g = sigm(g);
    outp[(size_t)bc*Hh + tid] = g*sp + (1.0f - g)*s_agg[tid];
  }
}

// ---------------------------------------------------------------------------
extern "C" void kernel_launch(void* const* d_in, const int* in_sizes, int n_in,
                              void* d_out, int out_size, void* d_ws, size_t ws_size,
                              hipStream_t stream) {
  const int* questions      = (const int*)d_in[0];
  const int* cand           = (const int*)d_in[1];
  const int* link_ents      = (const int*)d_in[2];
  const int* link_rels      = (const int*)d_in[3];
  const int* query_entities = (const int*)d_in[4];
  const int* rel_word_ids   = (const int*)d_in[5];
  const float* word_emb     = (const float*)d_in[6];
  const float* entity_table = (const float*)d_in[7];
  const float* W_ent = (const float*)d_in[8];
  const float* b_ent = (const float*)d_in[9];
  const float* Wih_f = (const float*)d_in[10];
  const float* Whh_f = (const float*)d_in[11];
  const float* b_lf  = (const float*)d_in[12];
  const float* Wih_b = (const float*)d_in[13];
  const float* Whh_b = (const float*)d_in[14];
  const float* b_lb  = (const float*)d_in[15];
  const float* att_w = (const float*)d_in[16];
  const float* att_b = (const float*)d_in[17];
  const float* W_qrel = (const float*)d_in[18];
  const float* b_qrel = (const float*)d_in[19];
  const float* W_cq   = (const float*)d_in[20];
  const float* b_cq   = (const float*)d_in[21];
  const float* W_kgp  = (const float*)d_in[22];
  const float* b_kgp  = (const float*)d_in[23];
  const float* W_kgg  = (const float*)d_in[24];
  const float* b_kgg  = (const float*)d_in[25];
  const float* W_sp   = (const float*)d_in[26];
  const float* b_sp   = (const float*)d_in[27];
  float* out = (float*)d_out;

  // workspace layout (floats)
  float* ws = (float*)d_ws;
  float* q_emb    = ws;                    // 16*20*64   = 20480
  float* rel_seq  = q_emb   + 20480;       // 500*10*64  = 320000
  float* rel_pool = rel_seq + 320000;      // 500*64     = 32000
  float* ent0     = rel_pool + 32000;      // 8192*64    = 524288
  float* ent1     = ent0    + 524288;      // 8192*64    = 524288
  float* qs       = ent1    + 524288;      // 8192*50    = 409600

  bilstm_kernel<<<Bq, 128, 0, stream>>>(questions, Bq, Tq, word_emb,
      Wih_f, Whh_f, b_lf, Wih_b, Whh_b, b_lb, q_emb, /*minlen=*/0);
  bilstm_kernel<<<Rr, 128, 0, stream>>>(rel_word_ids, Rr, TRr, word_emb,
      Wih_f, Whh_f, b_lf, Wih_b, Whh_b, b_lb, rel_seq, /*minlen=*/1);
  attnpool_kernel<<<Rr, 64, 0, stream>>>(rel_seq, rel_word_ids, att_w, att_b, rel_pool);
  ent0_kernel<<<BC/16, 128, 0, stream>>>(cand, entity_table, W_ent, b_ent, ent0);
  main_kernel<<<BC, 256, 0, stream>>>(questions, link_ents, link_rels,
      q_emb, rel_pool, ent0, W_qrel, b_qrel, W_cq, b_cq, ent1, qs);
  pass2_kernel<<<BC, 128, 0, stream>>>(link_ents, link_rels, query_entities,
      rel_pool, ent1, qs, W_kgp, b_kgp, W_kgg, b_kgg, W_sp, b_sp, out);
}